// TTTConv_3573412790638
// MI455X (gfx1250) — compile-verified
//
#include <hip/hip_runtime.h>
#include <cstdint>
#include <cstddef>

// Problem constants (match reference)
#define B_  4
#define S_  4096
#define H_  2048
#define K_  4
#define EPS_ 1e-6f

// Tiling
#define TS       32                 // tokens per block tile
#define NROWS    (TS + K_ - 1)      // 35 rows streamed per block (3 halo)
#define DEPTH    3                  // async prefetch depth (rows)
#define NTHREADS 256                // 8 wave32
#define PERTH    (H_ / NTHREADS)    // 8 channels per thread
#define RING     4                  // LDS ring slots (> DEPTH)

static_assert(H_ == NTHREADS * PERTH, "channel tiling");
static_assert(RING > DEPTH, "ring must exceed prefetch depth");

typedef float v4f __attribute__((ext_vector_type(4)));

// ---- CDNA5 async global->LDS copy path ------------------------------------
// GVS mode: mem_addr = SADDR(u64) + VADDR(i32) ; LDS dest offset in VDST vgpr.
__device__ __forceinline__ void async_ld_b128(unsigned lds_off, unsigned voff,
                                              uint64_t sbase) {
    asm volatile("global_load_async_to_lds_b128 %0, %1, %2"
                 :: "v"(lds_off), "v"(voff), "s"(sbase)
                 : "memory");
}
__device__ __forceinline__ void wait_async_le4() { asm volatile("s_wait_asynccnt 0x4" ::: "memory"); }
__device__ __forceinline__ void wait_async_le2() { asm volatile("s_wait_asynccnt 0x2" ::: "memory"); }
__device__ __forceinline__ void wait_async_le0() { asm volatile("s_wait_asynccnt 0x0" ::: "memory"); }
// ---------------------------------------------------------------------------

__global__ __launch_bounds__(NTHREADS, 1)
void tttconv_rmsnorm_dwconv(const float* __restrict__ hs,   // [B,S,H]
                            const float* __restrict__ nw,   // [H]
                            const float* __restrict__ cw,   // [H,1,K]
                            const float* __restrict__ cb,   // [H]
                            float* __restrict__ out)        // [B,S,H]
{
    __shared__ float ring[RING][H_];            // 32 KB async landing buffer
    __shared__ float red[2][NTHREADS / 32];     // cross-wave partials (dbl-buf)

    const int t    = threadIdx.x;
    const int wave = t >> 5;
    const int lane = t & 31;
    const int b    = blockIdx.y;
    const int s0   = blockIdx.x * TS;
    const int h0   = t * PERTH;

    // Fold norm weight into depthwise conv weights (per-thread channels).
    float a[PERTH][K_], bias[PERTH];
    #pragma unroll
    for (int j = 0; j < PERTH; ++j) {
        const float w = nw[h0 + j];
        bias[j] = cb[h0 + j];
        #pragma unroll
        for (int k = 0; k < K_; ++k)
            a[j][k] = w * cw[(h0 + j) * K_ + k];
    }

    const unsigned byteoff = (unsigned)(h0 * 4);   // this lane's 32B within a row

    auto issue_row = [&](int i) {
        int srow = s0 - (K_ - 1) + i;
        int sc   = srow < 0 ? 0 : srow;            // clamped addr; zeroed later
        uint64_t base = (uint64_t)(uintptr_t)(hs + ((size_t)b * S_ + sc) * H_);
        unsigned lds  = (unsigned)(uintptr_t)&ring[i & (RING - 1)][0] + byteoff;
        async_ld_b128(lds,       byteoff,       base);
        async_ld_b128(lds + 16u, byteoff + 16u, base);
    };

    float win[RING][PERTH];   // rolling 4-row normalized-input window (raw x)
    float invr[RING];         // rolling rsqrt(mean(x^2)+eps) per row

    // Prologue: fill the async pipeline.
    #pragma unroll
    for (int i = 0; i < DEPTH; ++i) issue_row(i);

    for (int i = 0; i < NROWS; ++i) {
        // Guarantee row i has landed (in-order async completion):
        // rows issued ahead of i: min(DEPTH-1, NROWS-1-i) -> 2 instrs each.
        if (i < NROWS - 2)       wait_async_le4();
        else if (i == NROWS - 2) wait_async_le2();
        else                     wait_async_le0();

        const int slot = i & (RING - 1);
        const int srow = s0 - (K_ - 1) + i;

        // Read back exactly the bytes this lane DMA'd.
        const v4f* lp = (const v4f*)&ring[slot][h0];
        v4f v0 = lp[0];
        v4f v1 = lp[1];
        if (srow < 0) {                      // causal left padding
            v4f z = {0.f, 0.f, 0.f, 0.f};
            v0 = z; v1 = z;
        }

        float p = 0.f;
        #pragma unroll
        for (int j = 0; j < 4; ++j) {
            win[slot][j]     = v0[j];
            win[slot][4 + j] = v1[j];
            p += v0[j] * v0[j] + v1[j] * v1[j];
        }

        // wave32 tree reduction, then 8 cross-wave partials.
        #pragma unroll
        for (int off = 16; off > 0; off >>= 1)
            p += __shfl_xor(p, off, 32);
        if (lane == 0) red[i & 1][wave] = p;
        __syncthreads();
        float tot = 0.f;
        #pragma unroll
        for (int wv = 0; wv < NTHREADS / 32; ++wv) tot += red[i & 1][wv];
        invr[slot] = rsqrtf(tot * (1.0f / (float)H_) + EPS_);

        // Keep the pipeline full (slot (i+DEPTH)&3 was consumed at iter i-1).
        if (i + DEPTH < NROWS) issue_row(i + DEPTH);

        // Emit output row once the 4-row window is populated.
        if (i >= K_ - 1) {
            const int sout = srow;           // = s0 + i - (K_-1)
            float yy[PERTH];
            #pragma unroll
            for (int j = 0; j < PERTH; ++j) {
                float acc = bias[j];
                #pragma unroll
                for (int k = 0; k < K_; ++k) {
                    const int r = (i - (K_ - 1) + k) & (RING - 1);
                    acc += a[j][k] * invr[r] * win[r][j];
                }
                yy[j] = acc;
            }
            v4f y0 = {yy[0], yy[1], yy[2], yy[3]};
            v4f y1 = {yy[4], yy[5], yy[6], yy[7]};
            v4f* op = (v4f*)(out + ((size_t)b * S_ + sout) * H_ + h0);
            __builtin_nontemporal_store(y0, op);       // streaming store
            __builtin_nontemporal_store(y1, op + 1);
        }
    }
}

extern "C" void kernel_launch(void* const* d_in, const int* in_sizes, int n_in,
                              void* d_out, int out_size, void* d_ws, size_t ws_size,
                              hipStream_t stream) {
    (void)in_sizes; (void)n_in; (void)out_size; (void)d_ws; (void)ws_size;
    const float* hs = (const float*)d_in[0];   // hidden_states [B,S,H]
    const float* nw = (const float*)d_in[1];   // norm_weight   [H]
    const float* cw = (const float*)d_in[2];   // conv_weight   [H,1,K]
    const float* cb = (const float*)d_in[3];   // conv_bias     [H]
    float* out = (float*)d_out;                // [B,S,H]

    dim3 grid(S_ / TS, B_);
    dim3 block(NTHREADS);
    hipLaunchKernelGGL(tttconv_rmsnorm_dwconv, grid, block, 0, stream,
                       hs, nw, cw, cb, out);
}